// Qatten_Weight_17454747091732
// MI455X (gfx1250) — compile-verified
//
#include <hip/hip_runtime.h>
#include <hip/hip_bf16.h>

// ---------------------------------------------------------------------------
// Qatten_Weight on MI455X (gfx1250).
// Math facts used:
//   * logits are constant across the agent axis -> softmax == 1/8 exactly,
//     so head_attend == sum_h p_h (== 0.5) broadcast over (BS, 8).
//   * v == V broadcast over (BS, 1).
//   * attend_mag_regs = 1e-3 * sum_h logit_h^2,
//     head_entropies[h] = -8 * p * log(p + 1e-8), p = 1/8.
//   * agent_qs / actions are never read by any output -> skip 96 MB of loads.
// The only FLOPs are diag(sel_w @ key_w^T) for a 4x64 pair -> done with
// v_wmma_f32_16x16x4_f32 (full fp32, 16 accumulating WMMAs over K=64).
// Everything else is a store-bound 37.7 MB fill (~1.6 us at 23.3 TB/s).
// ---------------------------------------------------------------------------

typedef __attribute__((ext_vector_type(2))) float v2f;
typedef __attribute__((ext_vector_type(8))) float v8f;

#define N_AGENTS 8
#define N_HEAD 4
#define EMBED_DIM 64
#define BS_TOTAL 1048576LL

// ---- Kernel A: one wave32. WMMA f32 dot products + scalar epilogue. -------
__global__ __launch_bounds__(32) void qatten_scalars_kernel(
    const float* __restrict__ sel_w,   // (4, 64)
    const float* __restrict__ key_w,   // (4, 64)
    const float* __restrict__ V,       // (1,)
    float* __restrict__ out,           // d_out (float)
    float* __restrict__ ws)            // d_ws scratch (float)
{
    const int lane = threadIdx.x;          // 0..31, EXEC all ones (WMMA req.)
    const int m    = lane & 15;            // row (A) / col (B) index
    const int khalf = (lane >> 4) * 2;     // lanes 0-15 -> K{0,1}; 16-31 -> K{2,3}

    // C/D accumulator: 16x16 f32 across 8 VGPRs per lane.
    v8f c = {};

    // A: 16x4 f32 slab of sel_w rows (rows 4..15 zero). VGPR0=K even, VGPR1=K odd.
    // B: 4x16 f32 slab of key_w^T (cols 4..15 zero). Same per-lane indexing since
    //    B[k, n] = key_w[n*64 + k] and n == lane&15 == m.
    for (int kb = 0; kb < EMBED_DIM / 4; ++kb) {
        const int k0 = kb * 4 + khalf;
        v2f a = {0.0f, 0.0f};
        v2f b = {0.0f, 0.0f};
        if (m < N_HEAD) {
            a.x = sel_w[m * EMBED_DIM + k0];
            a.y = sel_w[m * EMBED_DIM + k0 + 1];
            b.x = key_w[m * EMBED_DIM + k0];
            b.y = key_w[m * EMBED_DIM + k0 + 1];
        }
        // D = A*B + C ; emits v_wmma_f32_16x16x4_f32
        c = __builtin_amdgcn_wmma_f32_16x16x4_f32(
            /*neg_a=*/false, a, /*neg_b=*/false, b,
            /*c_mod=*/(short)0, c, /*reuse_a=*/false, /*reuse_b=*/false);
    }

    // Scatter accumulator to LDS so lane 0 can read diagonal entries.
    // C layout: element (M = r + 8*(lane>>4), N = lane&15) lives in VGPR r.
    // Diagonal (h,h), h<4  ->  lane h, VGPR h.
    __shared__ float cmat[32 * 8];
    #pragma unroll
    for (int r = 0; r < 8; ++r) cmat[lane * 8 + r] = c[r];
    __syncthreads();

    if (lane == 0) {
        const float scale = 0.125f;  // 1/sqrt(64)
        float hav = 0.0f;            // sum over heads of per-agent prob
        float reg = 0.0f;
        float ent[N_HEAD];
        #pragma unroll
        for (int h = 0; h < N_HEAD; ++h) {
            const float logit = cmat[h * 8 + h];
            reg += logit * logit;
            // softmax over N_AGENTS identical logits -> exactly 1/8
            const float xs = logit * scale;
            const float e  = __expf(xs - xs);            // == 1.0f
            const float p  = e / (8.0f * e);             // == 0.125f
            hav += p;
            ent[h] = -((float)N_AGENTS * p * __logf(p + 1e-8f));
        }
        const long long tail = BS_TOTAL * N_AGENTS + BS_TOTAL;
        out[tail] = 0.001f * reg;                        // attend_mag_regs
        #pragma unroll
        for (int h = 0; h < N_HEAD; ++h) out[tail + 1 + h] = ent[h];
        ws[0] = hav;                                     // head_attend value
        ws[1] = V[0];                                    // v value
    }
}

// ---- Kernel B: store-bound fill of head_attend (BS*8) and v (BS). ---------
__global__ __launch_bounds__(256) void qatten_fill_kernel(
    const float* __restrict__ ws,
    float4* __restrict__ out4,
    long long n4_head,     // float4 count of head_attend region
    long long n4_total)    // float4 count of head_attend + v regions
{
    const float hav = ws[0];
    const float vv  = ws[1];
    const float4 h4 = make_float4(hav, hav, hav, hav);
    const float4 v4 = make_float4(vv, vv, vv, vv);

    long long i = (long long)blockIdx.x * blockDim.x + threadIdx.x;
    const long long stride = (long long)gridDim.x * blockDim.x;
    for (; i < n4_total; i += stride) {
        out4[i] = (i < n4_head) ? h4 : v4;   // global_store_b128
    }
}

extern "C" void kernel_launch(void* const* d_in, const int* in_sizes, int n_in,
                              void* d_out, int out_size, void* d_ws, size_t ws_size,
                              hipStream_t stream) {
    // Input order: agent_qs(f32), actions(i64, unused), sel_w(f32), key_w(f32), V(f32)
    const float* sel_w = (const float*)d_in[2];
    const float* key_w = (const float*)d_in[3];
    const float* Vp    = (const float*)d_in[4];
    float* out = (float*)d_out;
    float* ws  = (float*)d_ws;

    // 1) tiny WMMA + scalar epilogue (also writes the 5 tail outputs)
    qatten_scalars_kernel<<<1, 32, 0, stream>>>(sel_w, key_w, Vp, out, ws);

    // 2) 37.7 MB constant fill: head_attend (BS*8 floats) then v (BS floats).
    //    BS*8 is divisible by 4 -> clean float4 split.
    const long long n4_head  = (BS_TOTAL * N_AGENTS) / 4;  // 2,097,152
    const long long n4_total = n4_head + BS_TOTAL / 4;     // 2,359,296
    const int threads = 256;
    const int blocks  = 4096;   // grid-stride; ~2.25 float4 per thread
    qatten_fill_kernel<<<blocks, threads, 0, stream>>>(ws, (float4*)out,
                                                       n4_head, n4_total);
}